// PointPillarsScatter_58737972740529
// MI455X (gfx1250) — compile-verified
//
#include <hip/hip_runtime.h>
#include <stdint.h>

// PointPillarsScatter for MI455X (gfx1250).
// Kernel 1: zero-fill 252MB output with B128 stores (RT hint -> most of the output
//           stays in the 192MB L2 so the scattered dword stores merge in L2).
// Kernel 2: per workgroup, TDM-load a contiguous 256-pillar x 64-ch f32 tile (64KB)
//           into LDS (tensor_load_to_lds / TENSORcnt), overlap the DMA with the
//           coords->index computation, then 8 waves scatter to channel-major BEV.

#define NXDIM 496
#define NYDIM 496
#define C_CH 64
#define BATCH 4
#define NPILLAR 12000
#define SPATIAL (NXDIM * NYDIM)        // 246016
#define TILE_P 256
#define TILE_ELEMS (TILE_P * C_CH)     // 16384 f32 = 64KB

typedef __attribute__((ext_vector_type(4))) unsigned int tdm_u32x4;
typedef __attribute__((ext_vector_type(4))) int tdm_i32x4;
typedef __attribute__((ext_vector_type(8))) int tdm_i32x8;

__global__ void bev_fill_zero(float4* __restrict__ out, unsigned int n4) {
    unsigned int i = blockIdx.x * blockDim.x + threadIdx.x;
    const unsigned int stride = gridDim.x * blockDim.x;
    const float4 z = make_float4(0.f, 0.f, 0.f, 0.f);
    for (; i < n4; i += stride)
        out[i] = z;   // global_store_b128, regular-temporal
}

__global__ void __launch_bounds__(256)
bev_scatter(const float* __restrict__ feat,
            const int* __restrict__ coords,
            float* __restrict__ out) {
    __shared__ float lds_feat[TILE_ELEMS];   // 64KB tile, filled by TDM
    __shared__ int   lds_idx[TILE_P];        // spatial index per pillar (-1 = drop)

    const int b   = blockIdx.y;
    const int p0  = blockIdx.x * TILE_P;
    const int tid = threadIdx.x;

    // ---- phase 0: issue TDM async tensor load FIRST (wave 0 only) -----------
    // Tile is fully contiguous in [B,P,C] row-major memory -> 1-D tile
    // descriptor. tensor_dim0 = remaining valid elements so the tail tile's
    // out-of-bounds reads return zero (ISA OOB rule). The DMA runs while all
    // waves compute per-pillar indices below.
    if (tid < 32) {  // TDM ignores EXEC; issue exactly once per workgroup
        const unsigned long long gaddr =
            (unsigned long long)(uintptr_t)(feat + ((long)b * NPILLAR + p0) * C_CH);
        const unsigned int lds_addr = (unsigned int)(uintptr_t)&lds_feat[0];
        const int remain = (NPILLAR - p0) * C_CH;
        const unsigned int nelem =
            (unsigned int)(remain < TILE_ELEMS ? remain : TILE_ELEMS);

        // D# group 0 (128b): [1:0]count=1 | [63:32]lds_addr |
        //                    [120:64]global_addr | [127:126]type=2
        tdm_u32x4 g0 = {
            1u,
            lds_addr,
            (unsigned int)(gaddr & 0xFFFFFFFFull),
            (unsigned int)((gaddr >> 32) & 0x1FFFFFFull) | (2u << 30)
        };
        // D# group 1 (256b): [17:16]data_size=2 (4B) | [79:48]tensor_dim0 |
        //   [111:80]tensor_dim1=1 | [127:112]tile_dim0 | [143:128]tile_dim1=0 |
        //   [207:160]tensor_dim0_stride
        tdm_i32x8 g1 = {
            (int)(2u << 16),                               // dword0
            (int)((nelem & 0xFFFFu) << 16),                // dword1: dim0[15:0]
            (int)((nelem >> 16) | (1u << 16)),             // dword2: dim0[31:16]|dim1
            (int)(((unsigned)TILE_ELEMS) << 16),           // dword3: tile_dim0
            0,                                             // dword4: tile_dim1/2 = 0
            (int)((unsigned)TILE_ELEMS),                   // dword5: dim0_stride lo
            0, 0
        };
        tdm_i32x4 g2 = {0, 0, 0, 0};
        tdm_i32x4 g3 = {0, 0, 0, 0};
#if __clang_major__ >= 23
        tdm_i32x8 g4 = {0, 0, 0, 0, 0, 0, 0, 0};
        __builtin_amdgcn_tensor_load_to_lds(g0, g1, g2, g3, g4, 0);
#else
        __builtin_amdgcn_tensor_load_to_lds(g0, g1, g2, g3, 0);
#endif
    }

    // ---- phase 1: per-pillar spatial index (overlaps the tensor DMA) --------
    {
        const int p = p0 + tid;
        int sidx = -1;
        if (p < NPILLAR) {
            const int* cp = coords + ((long)b * NPILLAR + p) * 3;
            const int x = cp[1];
            const int y = cp[2];
            if ((x + y > 0) && x >= 0 && x < NXDIM && y >= 0 && y < NYDIM)
                sidx = x * NYDIM + y;
        }
        lds_idx[tid] = sidx;
    }

    // ---- phase 2: wait for DMA, publish tile to all waves -------------------
    if (tid < 32)
        __builtin_amdgcn_s_wait_tensorcnt(0);   // s_wait_tensorcnt 0
    __syncthreads();

    // ---- phase 3: scatter LDS tile into channel-major BEV -------------------
    // lanes sweep channels (conflict-free LDS banks); stores are inherently
    // scattered dwords (channel stride = SPATIAL), merged in L2.
    const long out_b = (long)b * C_CH * SPATIAL;
    for (int i = tid; i < TILE_ELEMS; i += 256) {
        const int p_local = i >> 6;    // pillar within tile
        const int c       = i & 63;    // channel
        const int sidx    = lds_idx[p_local];
        if (sidx >= 0)
            out[out_b + (long)c * SPATIAL + sidx] = lds_feat[i];
    }
}

extern "C" void kernel_launch(void* const* d_in, const int* in_sizes, int n_in,
                              void* d_out, int out_size, void* d_ws, size_t ws_size,
                              hipStream_t stream) {
    const float* feat   = (const float*)d_in[0];   // (B, P, C) f32
    const int*   coords = (const int*)d_in[1];     // (B, P, 3) i32
    float*       out    = (float*)d_out;           // (B, C, NX, NY) f32

    // 1) zero-fill output (out_size = B*C*NX*NY = 62,980,096, divisible by 4)
    const unsigned int n4 = (unsigned int)(out_size / 4);
    bev_fill_zero<<<4096, 256, 0, stream>>>((float4*)out, n4);

    // 2) scatter (stream order guarantees the fill completes first)
    dim3 grid((NPILLAR + TILE_P - 1) / TILE_P, BATCH);
    bev_scatter<<<grid, 256, 0, stream>>>(feat, coords, out);
}